// GGCNLayer_46961172414534
// MI455X (gfx1250) — compile-verified
//
#include <hip/hip_runtime.h>

#define DDIM 128
#define NNODES 50000
#define NEDGES 500000
#define BN_EPS 1e-5f
#define AGG_EPS 1e-6f

typedef float v2f __attribute__((ext_vector_type(2)));
typedef float v4f __attribute__((ext_vector_type(4)));
typedef float v8f __attribute__((ext_vector_type(8)));

#define ND ((size_t)NNODES * DDIM)         // 6,400,000 floats
#define ED ((size_t)NEDGES * DDIM)         // 64,000,000 floats

// workspace layout (in floats)
#define OFF_AH    ((size_t)0)
#define OFF_BH    (OFF_AH + ND)
#define OFF_DH    (OFF_BH + ND)
#define OFF_EH    (OFF_DH + ND)
#define OFF_HNEW  (OFF_EH + ND)
#define OFF_SUMSH (OFF_HNEW + ND)
#define OFF_SUMS  (OFF_SUMSH + ND)
#define OFF_STATS (OFF_SUMS + ND)          // 1024 floats of stats
// stats sublayout: [0)h_sum [128)h_sumsq [256)e_sum [384)e_sumsq
//                  [512)h_mean [640)h_inv [768)e_mean [896)e_inv

#define NODE_TILES (NNODES / 16)           // 3125
#define EDGE_TILES (NEDGES / 16)           // 31250
#define TPB 8                              // row tiles per block (weight reuse factor)
#define NODE_BLOCKS ((NODE_TILES + TPB - 1) / TPB)   // 391
#define EDGE_BLOCKS ((EDGE_TILES + TPB - 1) / TPB)   // 3907
#define XS_STRIDE 132                      // padded LDS stride (floats) to avoid bank conflicts

static __device__ __forceinline__ void atomAddF32(float* p, float v) {
    __hip_atomic_fetch_add(p, v, __ATOMIC_RELAXED, __HIP_MEMORY_SCOPE_AGENT);
}

// ---------------------------------------------------------------------------
// 0. zero scratch accumulators
// ---------------------------------------------------------------------------
__global__ __launch_bounds__(256) void zero_kernel(float* __restrict__ p, long n4) {
    long i = (long)blockIdx.x * blockDim.x + threadIdx.x;
    if (i < n4) ((v4f*)p)[i] = (v4f){0.f, 0.f, 0.f, 0.f};
}

// ---------------------------------------------------------------------------
// 1. fused GEMM:  out = X @ W^T + b   for the 5 projections (FP32 WMMA)
//    one wave per 16-col tile, 8 waves/block; TPB row tiles per block with
//    B-fragments hoisted into registers (weights fetched once per block).
//    The edge GEMM (g==4) streams with non-temporal hints to protect L2.
// ---------------------------------------------------------------------------
__global__ __launch_bounds__(256) void gemm_kernel(
    const float* __restrict__ h, const float* __restrict__ e,
    const float* __restrict__ WA, const float* __restrict__ bA,
    const float* __restrict__ WB, const float* __restrict__ bB,
    const float* __restrict__ WC, const float* __restrict__ bC,
    const float* __restrict__ WD, const float* __restrict__ bD,
    const float* __restrict__ WE, const float* __restrict__ bE,
    float* __restrict__ Ah, float* __restrict__ Bh,
    float* __restrict__ Dh, float* __restrict__ Eh,
    float* __restrict__ Ce)
{
    __shared__ float xs[16 * XS_STRIDE];

    int bx = blockIdx.x;
    const float* X; const float* W; const float* bias; float* out;
    int tile0, limit; bool streaming;
    if (bx < 4 * NODE_BLOCKS) {
        int g = bx / NODE_BLOCKS;
        tile0 = (bx % NODE_BLOCKS) * TPB;
        limit = NODE_TILES; X = h; streaming = false;
        if (g == 0)      { W = WA; bias = bA; out = Ah; }
        else if (g == 1) { W = WB; bias = bB; out = Bh; }
        else if (g == 2) { W = WD; bias = bD; out = Dh; }
        else             { W = WE; bias = bE; out = Eh; }
    } else {
        tile0 = (bx - 4 * NODE_BLOCKS) * TPB;
        limit = EDGE_TILES; X = e; W = WC; bias = bC; out = Ce; streaming = true;
    }
    int ntiles = limit - tile0; if (ntiles > TPB) ntiles = TPB;

    const int wave = threadIdx.x >> 5;    // 0..7 : column tile
    const int lane = threadIdx.x & 31;
    const int hi   = lane >> 4;           // 0 or 1 (lane half)
    const int l16  = lane & 15;
    const int colbase = wave * 16;

    // hoist B fragments (W^T tile) into registers: 32 x v2f per lane
    v2f bfrag[32];
    {
        const float* wrow = W + (size_t)(colbase + l16) * DDIM;
#pragma unroll
        for (int kk = 0; kk < 32; ++kk) {
            v4f w4 = *(const v4f*)(wrow + kk * 4);
            bfrag[kk].x = hi ? w4.y : w4.x;
            bfrag[kk].y = hi ? w4.w : w4.z;
        }
    }
    const float bv = bias[colbase + l16];

    for (int t = 0; t < ntiles; ++t) {
        const long rowbase = (long)(tile0 + t) * 16;

        __syncthreads();  // previous tile's LDS reads complete
        for (int idx = threadIdx.x; idx < 16 * 32; idx += 256) {
            int row = idx >> 5;           // 32 float4 per row
            int c4  = (idx & 31) << 2;
            const v4f* sp = (const v4f*)(X + (rowbase + row) * DDIM + c4);
            v4f v = streaming ? __builtin_nontemporal_load(sp) : *sp;
            *(v4f*)&xs[row * XS_STRIDE + c4] = v;
        }
        __syncthreads();

        v8f acc = {0.f, 0.f, 0.f, 0.f, 0.f, 0.f, 0.f, 0.f};
#pragma unroll
        for (int k0 = 0; k0 < DDIM; k0 += 4) {
            // A fragment (16x4 f32): VGPR0 = K{0,2}, VGPR1 = K{1,3} per lane half
            v2f a = *(const v2f*)&xs[l16 * XS_STRIDE + k0 + 2 * hi];
            acc = __builtin_amdgcn_wmma_f32_16x16x4_f32(
                /*neg_a=*/false, a, /*neg_b=*/false, bfrag[k0 >> 2],
                /*c_mod=*/(short)0, acc, /*reuse_a=*/false, /*reuse_b=*/false);
        }

        float* orow = out + (rowbase + hi * 8) * DDIM + colbase + l16;
#pragma unroll
        for (int r = 0; r < 8; ++r) {
            float val = acc[r] + bv;
            if (streaming) __builtin_nontemporal_store(val, orow + (size_t)r * DDIM);
            else           orow[(size_t)r * DDIM] = val;
        }
    }
}

// ---------------------------------------------------------------------------
// 2. edge update + gated scatter:  e_new = Dh[src]+Eh[dst]+Ce ; sigma=sig(e_new)
//    sum_sh[dst] += Bh[src]*sigma ; sum_s[dst] += sigma
//    Ce is staged in-place in the e-half of d_out (non-temporal RMW);
//    gathers + atomics target L2-resident node buffers.
// ---------------------------------------------------------------------------
__global__ __launch_bounds__(256) void edge_kernel(
    const int* __restrict__ src, const int* __restrict__ dst,
    const float* __restrict__ Bh, const float* __restrict__ Dh,
    const float* __restrict__ Eh,
    float* e_stage,                       // in: Ce, out: e_new  (aliases d_out)
    float* __restrict__ sum_sh, float* __restrict__ sum_s)
{
    long idx = (long)blockIdx.x * blockDim.x + threadIdx.x;
    long edge = idx >> 5;
    if (edge >= NEDGES) return;
    int col = (int)(idx & 31) << 2;

    int s = src[edge], d = dst[edge];
    v4f dh = *(const v4f*)(Dh + (size_t)s * DDIM + col);
    v4f eh = *(const v4f*)(Eh + (size_t)d * DDIM + col);
    v4f* ep = (v4f*)(e_stage + (size_t)edge * DDIM + col);
    v4f en = __builtin_nontemporal_load(ep);
#pragma unroll
    for (int i = 0; i < 4; ++i) en[i] += dh[i] + eh[i];
    __builtin_nontemporal_store(en, ep);

    v4f sg;
#pragma unroll
    for (int i = 0; i < 4; ++i) sg[i] = 1.0f / (1.0f + __expf(-en[i]));

    v4f bh = *(const v4f*)(Bh + (size_t)s * DDIM + col);
    float* p1 = sum_sh + (size_t)d * DDIM + col;
    float* p2 = sum_s  + (size_t)d * DDIM + col;
#pragma unroll
    for (int i = 0; i < 4; ++i) {
        atomAddF32(p1 + i, bh[i] * sg[i]);
        atomAddF32(p2 + i, sg[i]);
    }
}

// ---------------------------------------------------------------------------
// 3. h_new = Ah + sum_sh / (sum_s + eps)
// ---------------------------------------------------------------------------
__global__ __launch_bounds__(256) void hnew_kernel(
    const float* __restrict__ Ah, const float* __restrict__ sum_sh,
    const float* __restrict__ sum_s, float* __restrict__ h_new)
{
    long i = (long)blockIdx.x * blockDim.x + threadIdx.x;
    if (i >= (long)(ND / 4)) return;
    v4f a  = ((const v4f*)Ah)[i];
    v4f sh = ((const v4f*)sum_sh)[i];
    v4f ss = ((const v4f*)sum_s)[i];
    v4f r;
#pragma unroll
    for (int k = 0; k < 4; ++k) r[k] = a[k] + sh[k] / (ss[k] + AGG_EPS);
    ((v4f*)h_new)[i] = r;
}

// ---------------------------------------------------------------------------
// 4. per-column sum / sum-of-squares (batch-norm statistics)
// ---------------------------------------------------------------------------
__global__ __launch_bounds__(256) void colstats_kernel(
    const float* __restrict__ x, long rows, int nt,
    float* __restrict__ sum, float* __restrict__ sumsq)
{
    int col = threadIdx.x & (DDIM - 1);
    int rl  = threadIdx.x >> 7;          // two row-lanes per block
    float s = 0.f, s2 = 0.f;
    for (long r = (long)blockIdx.x * 2 + rl; r < rows; r += (long)gridDim.x * 2) {
        const float* p = x + r * DDIM + col;
        float v = nt ? __builtin_nontemporal_load(p) : *p;
        s += v; s2 += v * v;
    }
    atomAddF32(sum + col, s);
    atomAddF32(sumsq + col, s2);
}

// ---------------------------------------------------------------------------
// 5. finalize BN statistics: mean + inverse std
// ---------------------------------------------------------------------------
__global__ __launch_bounds__(256) void bnstats_kernel(float* stats)
{
    int c = threadIdx.x;
    if (c < 128) {
        float m = stats[c] * (1.0f / NNODES);
        float v = stats[128 + c] * (1.0f / NNODES) - m * m;
        stats[512 + c] = m;
        stats[640 + c] = rsqrtf(v + BN_EPS);
    } else {
        int cc = c - 128;
        float m = stats[256 + cc] * (1.0f / NEDGES);
        float v = stats[384 + cc] * (1.0f / NEDGES) - m * m;
        stats[768 + cc] = m;
        stats[896 + cc] = rsqrtf(v + BN_EPS);
    }
}

// ---------------------------------------------------------------------------
// 6. out = residual + relu(bn(xnew))      (xnew/out alias for e; nt=streaming)
// ---------------------------------------------------------------------------
__global__ __launch_bounds__(256) void bnout_kernel(
    const float* __restrict__ resid, const float* xnew,
    const float* __restrict__ mean, const float* __restrict__ inv,
    const float* __restrict__ gamma, const float* __restrict__ beta,
    float* out, long total4, int nt)
{
    long i = (long)blockIdx.x * blockDim.x + threadIdx.x;
    if (i >= total4) return;
    int col = (int)(i & (DDIM / 4 - 1)) << 2;
    const v4f* pn = (const v4f*)xnew + i;
    const v4f* pr = (const v4f*)resid + i;
    v4f xn = nt ? __builtin_nontemporal_load(pn) : *pn;
    v4f xr = nt ? __builtin_nontemporal_load(pr) : *pr;
    v4f o;
#pragma unroll
    for (int k = 0; k < 4; ++k) {
        float t = (xn[k] - mean[col + k]) * inv[col + k] * gamma[col + k] + beta[col + k];
        o[k] = xr[k] + fmaxf(t, 0.f);
    }
    if (nt) __builtin_nontemporal_store(o, (v4f*)out + i);
    else    ((v4f*)out)[i] = o;
}

// ---------------------------------------------------------------------------
extern "C" void kernel_launch(void* const* d_in, const int* in_sizes, int n_in,
                              void* d_out, int out_size, void* d_ws, size_t ws_size,
                              hipStream_t stream) {
    const float* h    = (const float*)d_in[0];
    const float* e    = (const float*)d_in[1];
    const int*   src  = (const int*)d_in[2];
    const int*   dst  = (const int*)d_in[3];
    const float* WA = (const float*)d_in[4];  const float* bA = (const float*)d_in[5];
    const float* WB = (const float*)d_in[6];  const float* bB = (const float*)d_in[7];
    const float* WC = (const float*)d_in[8];  const float* bC = (const float*)d_in[9];
    const float* WD = (const float*)d_in[10]; const float* bD = (const float*)d_in[11];
    const float* WE = (const float*)d_in[12]; const float* bE = (const float*)d_in[13];
    const float* gamma_h = (const float*)d_in[14]; const float* beta_h = (const float*)d_in[15];
    const float* gamma_e = (const float*)d_in[16]; const float* beta_e = (const float*)d_in[17];

    float* ws   = (float*)d_ws;
    float* Ah   = ws + OFF_AH;
    float* Bh   = ws + OFF_BH;
    float* Dh   = ws + OFF_DH;
    float* Eh   = ws + OFF_EH;
    float* hnew = ws + OFF_HNEW;
    float* ssh  = ws + OFF_SUMSH;
    float* ss   = ws + OFF_SUMS;
    float* stats= ws + OFF_STATS;

    float* out_h = (float*)d_out;            // [NNODES*DDIM]
    float* out_e = out_h + ND;               // [NEDGES*DDIM] ; also Ce / e_new staging

    // 0. zero sum buffers + stats (contiguous: ssh, ss, stats)
    {
        long n4 = (long)(2 * ND + 1024) / 4;
        int blocks = (int)((n4 + 255) / 256);
        zero_kernel<<<blocks, 256, 0, stream>>>(ssh, n4);
    }
    // 1. all five GEMMs (Ce staged into out_e)
    gemm_kernel<<<4 * NODE_BLOCKS + EDGE_BLOCKS, 256, 0, stream>>>(
        h, e, WA, bA, WB, bB, WC, bC, WD, bD, WE, bE, Ah, Bh, Dh, Eh, out_e);
    // 2. edge update + gated scatter
    {
        long threads = (long)NEDGES * 32;
        edge_kernel<<<(int)((threads + 255) / 256), 256, 0, stream>>>(
            src, dst, Bh, Dh, Eh, out_e, ssh, ss);
    }
    // 3. h_new
    hnew_kernel<<<(int)((ND / 4 + 255) / 256), 256, 0, stream>>>(Ah, ssh, ss, hnew);
    // 4. BN statistics
    colstats_kernel<<<512, 256, 0, stream>>>(hnew, (long)NNODES, 0, stats + 0, stats + 128);
    colstats_kernel<<<2048, 256, 0, stream>>>(out_e, (long)NEDGES, 1, stats + 256, stats + 384);
    // 5. finalize stats
    bnstats_kernel<<<1, 256, 0, stream>>>(stats);
    // 6. outputs
    bnout_kernel<<<(int)((ND / 4 + 255) / 256), 256, 0, stream>>>(
        h, hnew, stats + 512, stats + 640, gamma_h, beta_h, out_h, (long)(ND / 4), 0);
    bnout_kernel<<<(int)((ED / 4 + 255) / 256), 256, 0, stream>>>(
        e, out_e, stats + 768, stats + 896, gamma_e, beta_e, out_e, (long)(ED / 4), 1);
}